// CircularConvLSTM_85521388798345
// MI455X (gfx1250) — compile-verified
//
#include <hip/hip_runtime.h>

// ---------------------------------------------------------------------------
// Circular ConvLSTM, 2 layers, fp32, CDNA5 (gfx1250).
// Gate GEMMs on the matrix pipe via V_WMMA_F32_16X16X4_F32 (exact fp32).
// One wave owns all 4 gate tiles (4 accumulators) for a 16ch x 16col tile.
// The reduction runs in rolled groups of 5 wmma-chunks (20 K values): the 10
// per-lane im2col addresses are materialized as pointers bumped by a uniform
// stride, so the group body is only loads + wmmas (no div/mod/select remat),
// and stays tiny in I$. A-fragment loads use immediate offsets (gate stride
// fits the 24-bit field). LSTM pointwise is fully register-resident.
// h ping-pongs between d_ws and the h_states region of d_out; c is in-place.
// ---------------------------------------------------------------------------

typedef float v2f __attribute__((ext_vector_type(2)));
typedef float v8f __attribute__((ext_vector_type(8)));

#define CB   32    // batch
#define CT   96    // time steps
#define CCIN 32    // layer-0 input channels
#define CP   168   // positions
#define CH   96    // hidden channels
#define CKK  5     // conv kernel size
#define CKHH (CH * CKK)          // 480  hidden-conv reduction length
#define CNP  (CB * CP)           // 5376 gemm columns
#define CBHP (CB * CH * CP)      // 516096 elems per state tensor

__global__ void zero_f32_kernel(float* __restrict__ p, int n) {
  int i = blockIdx.x * 256 + threadIdx.x;
  if (i < n) p[i] = 0.0f;
}

// KIH = input-conv reduction length (160 for layer 0, 480 for layer 1).
template <int KIH>
__global__ __launch_bounds__(128) void convlstm_step_kernel(
    const float* __restrict__ xin, int x_bstride,     // layer input
    const float* __restrict__ hprev,                  // [B, H, P]
    const float* __restrict__ w_ih,                   // [4H, Cin_l, K]
    const float* __restrict__ w_hh,                   // [4H, H, K]
    const float* __restrict__ b_ih,                   // [4H]
    const float* __restrict__ b_hh,                   // [4H]
    float* __restrict__ cbuf,                         // [B, H, P] in-place
    float* __restrict__ hnext,                        // [B, H, P]
    float* __restrict__ yout,                         // outputs base or null
    int t) {
  const int lane = threadIdx.x & 31;
  const int wv   = threadIdx.x >> 5;   // 4 waves, 4 independent column tiles
  const int m    = lane & 15;
  const int half = lane >> 4;
  const int klo  = half << 1;          // K-offset inside a 4-chunk: 0 or 2

  // GEMM column n -> (batch, position)
  const int ntile = blockIdx.x * 4 + wv;
  const int n = ntile * 16 + m;
  const int b = n / CP;
  const int p = n - b * CP;
  const int ch_tile = blockIdx.y * 16;

  // circularly wrapped positions for the 5 taps (only negative wrap possible)
  int q[5];
#pragma unroll
  for (int kk = 0; kk < 5; ++kk) {
    int pp = p + kk - 4;
    q[kk] = (pp < 0) ? pp + CP : pp;
  }

  // loop-invariant B element offsets: reduction index pattern repeats per 20
  int offs[5][2];
#pragma unroll
  for (int u = 0; u < 5; ++u) {
#pragma unroll
    for (int s = 0; s < 2; ++s) {
      const int j  = 4 * u + klo + s;     // 0..19
      const int ci = j / 5;
      offs[u][s] = ci * CP + q[j - ci * 5];
    }
  }

  // per-lane base pointers
  const float* __restrict__ xb = xin + (size_t)b * x_bstride;
  const float* __restrict__ hb = hprev + (size_t)b * (CH * CP);

  // A row pointers; per-gate stride handled with immediate load offsets
  constexpr int GIH = CH * KIH;   // elems between gates in w_ih
  constexpr int GHH = CH * CKHH;  // elems between gates in w_hh
  const float* __restrict__ wg = w_ih + (size_t)(ch_tile + m) * KIH + klo;
  const float* __restrict__ vg = w_hh + (size_t)(ch_tile + m) * CKHH + klo;

  // accumulators (one per gate), initialized with the per-row bias
  v8f acc0, acc1, acc2, acc3;
#pragma unroll
  for (int r = 0; r < 8; ++r) {
    const int ch = ch_tile + half * 8 + r;   // C-layout row for VGPR r
    acc0[r] = b_ih[ch]          + b_hh[ch];
    acc1[r] = b_ih[CH + ch]     + b_hh[CH + ch];
    acc2[r] = b_ih[2 * CH + ch] + b_hh[2 * CH + ch];
    acc3[r] = b_ih[3 * CH + ch] + b_hh[3 * CH + ch];
  }

  // materialized B-fragment pointers (10), bumped by a uniform group stride
  const float* pb[5][2];
#pragma unroll
  for (int u = 0; u < 5; ++u) {
    pb[u][0] = xb + offs[u][0];
    pb[u][1] = xb + offs[u][1];
  }

  // ---- W_ih @ im2col(x): rolled groups of 5 wmma-chunks (20 K = 4 channels)
#pragma unroll 1
  for (int g = 0; g < KIH / 20; ++g) {
#pragma unroll
    for (int u = 0; u < 5; ++u) {
      const v2f bf = { *pb[u][0], *pb[u][1] };
      const v2f a0 = *(const v2f*)(wg + u * 4);
      const v2f a1 = *(const v2f*)(wg + GIH + u * 4);
      const v2f a2 = *(const v2f*)(wg + 2 * GIH + u * 4);
      const v2f a3 = *(const v2f*)(wg + 3 * GIH + u * 4);
      acc0 = __builtin_amdgcn_wmma_f32_16x16x4_f32(false, a0, false, bf, (short)0, acc0, false, false);
      acc1 = __builtin_amdgcn_wmma_f32_16x16x4_f32(false, a1, false, bf, (short)0, acc1, false, false);
      acc2 = __builtin_amdgcn_wmma_f32_16x16x4_f32(false, a2, false, bf, (short)0, acc2, false, false);
      acc3 = __builtin_amdgcn_wmma_f32_16x16x4_f32(false, a3, false, bf, (short)0, acc3, false, false);
    }
#pragma unroll
    for (int u = 0; u < 5; ++u) {
      pb[u][0] += 4 * CP;   // 20 reduction steps == 4 channels
      pb[u][1] += 4 * CP;
    }
    wg += 20;
  }

  // ---- W_hh @ im2col(h_prev): same structure, pointers rebased onto h ----
#pragma unroll
  for (int u = 0; u < 5; ++u) {
    pb[u][0] = hb + offs[u][0];
    pb[u][1] = hb + offs[u][1];
  }
#pragma unroll 1
  for (int g = 0; g < CKHH / 20; ++g) {
#pragma unroll
    for (int u = 0; u < 5; ++u) {
      const v2f bf = { *pb[u][0], *pb[u][1] };
      const v2f a0 = *(const v2f*)(vg + u * 4);
      const v2f a1 = *(const v2f*)(vg + GHH + u * 4);
      const v2f a2 = *(const v2f*)(vg + 2 * GHH + u * 4);
      const v2f a3 = *(const v2f*)(vg + 3 * GHH + u * 4);
      acc0 = __builtin_amdgcn_wmma_f32_16x16x4_f32(false, a0, false, bf, (short)0, acc0, false, false);
      acc1 = __builtin_amdgcn_wmma_f32_16x16x4_f32(false, a1, false, bf, (short)0, acc1, false, false);
      acc2 = __builtin_amdgcn_wmma_f32_16x16x4_f32(false, a2, false, bf, (short)0, acc2, false, false);
      acc3 = __builtin_amdgcn_wmma_f32_16x16x4_f32(false, a3, false, bf, (short)0, acc3, false, false);
    }
#pragma unroll
    for (int u = 0; u < 5; ++u) {
      pb[u][0] += 4 * CP;
      pb[u][1] += 4 * CP;
    }
    vg += 20;
  }

  // ---- LSTM pointwise, entirely in registers (i,f,o,g share (M,N) per lane)
#pragma unroll
  for (int r = 0; r < 8; ++r) {
    const int ch = ch_tile + half * 8 + r;
    const float iv = 1.0f / (1.0f + __expf(-acc0[r]));
    const float fv = 1.0f / (1.0f + __expf(-acc1[r]));
    const float ov = 1.0f / (1.0f + __expf(-acc2[r]));
    const float gv = tanhf(acc3[r]);

    const int idx = (b * CH + ch) * CP + p;
    const float cnew = fv * cbuf[idx] + iv * gv;
    cbuf[idx] = cnew;
    const float hv = ov * tanhf(cnew);
    hnext[idx] = hv;
    if (yout) yout[((size_t)(b * CT + t) * CH + ch) * CP + p] = hv;
  }
}

extern "C" void kernel_launch(void* const* d_in, const int* in_sizes, int n_in,
                              void* d_out, int out_size, void* d_ws, size_t ws_size,
                              hipStream_t stream) {
  (void)in_sizes; (void)n_in; (void)out_size; (void)ws_size;

  const float* x     = (const float*)d_in[0];  // [B,T,Cin,P]
  const float* w_ih0 = (const float*)d_in[1];
  const float* b_ih0 = (const float*)d_in[2];
  const float* w_hh0 = (const float*)d_in[3];
  const float* b_hh0 = (const float*)d_in[4];
  const float* w_ih1 = (const float*)d_in[5];
  const float* b_ih1 = (const float*)d_in[6];
  const float* w_hh1 = (const float*)d_in[7];
  const float* b_hh1 = (const float*)d_in[8];

  float* out = (float*)d_out;
  const size_t outN = (size_t)CB * CT * CH * CP;  // outputs [B,T,H,P]
  float* hstates = out + outN;                    // h_states [2,B,H,P]
  float* h0o = hstates;
  float* h1o = hstates + CBHP;
  float* cst = hstates + 2 * (size_t)CBHP;        // c_states [2,B,H,P]
  float* c0  = cst;
  float* c1  = cst + CBHP;

  float* h0w = (float*)d_ws;                      // ping buffers for h
  float* h1w = h0w + CBHP;

  // zero h_states + c_states (t=0 reads h from the d_out region; c in-place)
  const int zn = 4 * CBHP;
  zero_f32_kernel<<<(zn + 255) / 256, 256, 0, stream>>>(hstates, zn);

  const dim3 grid(CNP / 64, CH / 16);  // 84 x 6 blocks, 4 waves/block
  for (int t = 0; t < CT; ++t) {
    // parity chosen so the final write (t = 95, odd) lands in d_out h_states
    float* r0 = (t & 1) ? h0w : h0o;
    float* w0 = (t & 1) ? h0o : h0w;
    float* r1 = (t & 1) ? h1w : h1o;
    float* w1 = (t & 1) ? h1o : h1w;

    convlstm_step_kernel<CCIN * CKK><<<grid, 128, 0, stream>>>(
        x + (size_t)t * CCIN * CP, CT * CCIN * CP, r0,
        w_ih0, w_hh0, b_ih0, b_hh0, c0, w0, nullptr, t);

    convlstm_step_kernel<CH * CKK><<<grid, 128, 0, stream>>>(
        w0, CH * CP, r1,
        w_ih1, w_hh1, b_ih1, b_hh1, c1, w1, out, t);
  }
}